// OneHot_fc_light_11613591568528
// MI455X (gfx1250) — compile-verified
//
#include <hip/hip_runtime.h>

typedef __attribute__((ext_vector_type(2))) float v2f;
typedef __attribute__((ext_vector_type(4))) float v4f;
typedef __attribute__((ext_vector_type(8))) float v8f;

#define DIM 128
#define WAVES_PER_BLOCK 4
#define THREADS (WAVES_PER_BLOCK * 32)
#define ROWS_PER_WAVE 16
#define ROWS_PER_BLOCK (ROWS_PER_WAVE * WAVES_PER_BLOCK)  // 64

// Fused: out = relu(W1[x] + b1) @ W2 + b2
// One wave per 16-row M-tile, f32 WMMA 16x16x4.
// Key points:
//  - W1 row gathers are issued BEFORE the W2->LDS staging + barrier so the
//    two HBM-latency events overlap (kernel is latency/bandwidth bound).
//  - W2 staged once per block into 64KB LDS with an XOR-16 swizzle keyed on
//    (k>>1)&1 so the two lane-halves of each B-fragment ds load hit disjoint
//    16-bank groups (conflict-free).
//  - ks-outer / nt-inner loop reuses each A fragment across 8 WMMAs; 8
//    independent v8f accumulator chains.
__global__ __launch_bounds__(THREADS) void onehot_fc_wmma_kernel(
    const int* __restrict__ x, const float* __restrict__ W1,
    const float* __restrict__ b1, const float* __restrict__ W2,
    const float* __restrict__ b2, float* __restrict__ out) {
  __shared__ float lds_w2[DIM * DIM];  // element (k,c) at k*128 + (c ^ (((k>>1)&1)<<4))

  const int t = threadIdx.x;
  const int wave = t >> 5;
  const int lane = t & 31;
  const int g = lane >> 4;   // lane half: selects K sub-pair / M offset
  const int lm = lane & 15;  // M (for A) / N (for B,C,D) within the 16-tile
  const int m0 = blockIdx.x * ROWS_PER_BLOCK + wave * ROWS_PER_WAVE;

  // ---- issue the random-row gather FIRST so it overlaps W2 staging ----
  // A-fragment layout: lane holds A[lm][4*ks + 2*g + {0,1}], ks = 0..31.
  const int row = m0 + lm;
  const int idx = x[row];
  const float* __restrict__ wrow = W1 + (size_t)idx * DIM;

  v2f wraw[32];
#pragma unroll
  for (int ks = 0; ks < 32; ++ks) {
    wraw[ks] = *(const v2f*)(wrow + ks * 4 + g * 2);
  }

  // ---- cooperative stage of W2 into LDS (float4, coalesced, swizzled) ----
  {
    const v4f* __restrict__ w2v = (const v4f*)W2;
#pragma unroll
    for (int it = 0; it < (DIM * DIM / 4) / THREADS; ++it) {
      const int v = it * THREADS + t;      // float4 index
      const int r = v >> 5;                // (v*4) / 128
      const int col4 = (v & 31) << 2;      // (v*4) % 128, 4-aligned
      const int scol = col4 ^ (((r >> 1) & 1) << 4);  // swizzle keeps 4-alignment
      v4f d = w2v[v];
      *(v4f*)&lds_w2[r * DIM + scol] = d;
    }
  }
  __syncthreads();

  // ---- finish A fragments: + b1, ReLU (b1 is 512B, cache-hot) ----
  v2f afrag[32];
#pragma unroll
  for (int ks = 0; ks < 32; ++ks) {
    const int c = ks * 4 + g * 2;
    v2f bb = *(const v2f*)(b1 + c);
    v2f a;
    a.x = fmaxf(wraw[ks].x + bb.x, 0.0f);
    a.y = fmaxf(wraw[ks].y + bb.y, 0.0f);
    afrag[ks] = a;
  }

  // ---- 16x128 output tile: 8 N-tiles accumulated over K=128 in steps of 4 ----
  v8f acc[8] = {};

  const int sw = g << 4;  // per-lane swizzle term ((kk>>1)&1 == g for kk=4ks+2g)
#pragma unroll 4
  for (int ks = 0; ks < 32; ++ks) {
    const int kk = ks * 4 + g * 2;  // B rows: kk (b.x) and kk+1 (b.y)
    const v2f a = afrag[ks];
#pragma unroll
    for (int nt = 0; nt < 8; ++nt) {
      const int scol = (nt * 16 + lm) ^ sw;
      v2f b;
      b.x = lds_w2[kk * DIM + scol];
      b.y = lds_w2[(kk + 1) * DIM + scol];
      acc[nt] = __builtin_amdgcn_wmma_f32_16x16x4_f32(
          false, a, false, b, (short)0, acc[nt], false, false);
    }
  }

  // ---- epilogue: + b2, store. D layout: VGPR v -> row v + 8*g, col lm ----
#pragma unroll
  for (int nt = 0; nt < 8; ++nt) {
    const int col = nt * 16 + lm;
    const float bias2 = b2[col];
#pragma unroll
    for (int vv = 0; vv < 8; ++vv) {
      const int r = m0 + vv + 8 * g;
      out[r * DIM + col] = acc[nt][vv] + bias2;
    }
  }
}

extern "C" void kernel_launch(void* const* d_in, const int* in_sizes, int n_in,
                              void* d_out, int out_size, void* d_ws, size_t ws_size,
                              hipStream_t stream) {
  (void)n_in; (void)out_size; (void)d_ws; (void)ws_size;
  const int* x = (const int*)d_in[0];          // [BATCH] int32
  const float* W1 = (const float*)d_in[1];     // [VOCAB, 128] f32
  const float* b1 = (const float*)d_in[2];     // [128] f32
  const float* W2 = (const float*)d_in[3];     // [128, 128] f32
  const float* b2 = (const float*)d_in[4];     // [128] f32
  float* out = (float*)d_out;                  // [BATCH, 128] f32

  const int batch = in_sizes[0];               // 4096
  dim3 grid(batch / ROWS_PER_BLOCK);           // 64 blocks of 128 threads (4 waves)
  onehot_fc_wmma_kernel<<<grid, THREADS, 0, stream>>>(x, W1, b1, W2, b2, out);
}